// ELKEncoder_64613488001615
// MI455X (gfx1250) — compile-verified
//
#include <hip/hip_runtime.h>
#include <math.h>

typedef float v2f __attribute__((ext_vector_type(2)));
typedef float v4f __attribute__((ext_vector_type(4)));
typedef float v8f __attribute__((ext_vector_type(8)));

#define CCH 64      // channels
#define KNB 27      // kernel offsets / neighbors
#define LDA 68      // LDS stride (floats) for A tile: conflict-free b64 frag reads
#define BSTEP 168   // B pairs per K-step group (16 mod-64-bank skew for stores)
#define BHSEL 80    // B pairs per hsel group (32-bank skew between lane halves)

// ---------------------------------------------------------------------------
// K2: local = sum_k F[nbr[:,k]] @ W_conv[k]   (the 66 GFLOP contraction)
// Block = 256 threads = 8 waves; block computes 256 rows x 64 cols.
// Each wave: TWO 16-row fragments (32 rows) x 4 col tiles -> 8 accumulators;
// every B fragment read from LDS feeds 2 WMMAs (0.75 ds_load per v_wmma).
// B is staged in fragment-major order so each fragment is one b64 read with
// a compile-time immediate offset.  Static LDS: 69632 + 21504 = 91136 B
// (gfx1250 WGP allows up to 320 KB per workgroup).
// ---------------------------------------------------------------------------
__global__ __launch_bounds__(256) void elk_conv_wmma(
    const float* __restrict__ F, const int* __restrict__ nbr,
    const float* __restrict__ Wconv, float* __restrict__ outLocal, int N)
{
    __shared__ float sA[256 * LDA];        // 69632 B
    __shared__ float sB[2 * 16 * BSTEP];   // 21504 B (fragment-major W_conv[k])

    const int tid  = threadIdx.x;
    const int lane = tid & 31;
    const int wave = tid >> 5;
    const long base = (long)blockIdx.x * 256;

    v8f acc0[4], acc1[4];
    #pragma unroll
    for (int t = 0; t < 4; ++t)
        #pragma unroll
        for (int j = 0; j < 8; ++j) { acc0[t][j] = 0.0f; acc1[t][j] = 0.0f; }

    const int m    = lane & 15;      // row (A) / col (B) within tile
    const int hsel = lane >> 4;      // 0 -> K pair {0,1}, 1 -> K pair {2,3}

    const float* aBase0 = &sA[(wave * 32 + m) * LDA + hsel * 2];
    const float* aBase1 = aBase0 + 16 * LDA;
    const float* bBase  = &sB[2 * (hsel * BHSEL + m)];

    #pragma unroll 1
    for (int k = 0; k < KNB; ++k) {
        // ---- stage gathered A tile: 256 rows x 64 ch (1 thread per row) ----
        {
            const long grow = base + tid;
            v4f* dst = (v4f*)&sA[tid * LDA];
            if (grow < (long)N) {
                const int idx = nbr[grow * KNB + k];
                const v4f* src = (const v4f*)(F + (long)idx * CCH);
                #pragma unroll
                for (int q = 0; q < 16; ++q) dst[q] = src[q];
            } else {
                v4f z = {0.f, 0.f, 0.f, 0.f};
                #pragma unroll
                for (int q = 0; q < 16; ++q) dst[q] = z;
            }
        }
        // ---- stage B = W_conv[k] (64x64) in fragment-major order ----
        // element (r = k-chan, c = col) -> pair (r>>2)*BSTEP + ((r>>1)&1)*BHSEL + c,
        // float slot 2*pair + (r&1): lane fragment {B[ks][c],B[ks+1][c]} is one b64.
        {
            const int r     = tid & 63;
            const int cq    = (tid >> 6) * 16;
            const float* src = Wconv + ((long)k * CCH + r) * CCH + cq;
            const int basep = (r >> 2) * BSTEP + ((r >> 1) & 1) * BHSEL + cq;
            const int bit   = r & 1;
            #pragma unroll
            for (int q = 0; q < 4; ++q) {
                v4f v = *(const v4f*)(src + q * 4);
                #pragma unroll
                for (int j = 0; j < 4; ++j)
                    sB[2 * (basep + q * 4 + j) + bit] = v[j];
            }
        }
        __syncthreads();

        // ---- 16 K-steps: 2 A b64 + 4 B b64 -> 8 v_wmma per step ----
        #pragma unroll
        for (int step = 0; step < 16; ++step) {
            v2f a0 = *(const v2f*)(aBase0 + step * 4);
            v2f a1 = *(const v2f*)(aBase1 + step * 4);
            #pragma unroll
            for (int t = 0; t < 4; ++t) {
                v2f b = *(const v2f*)(bBase + 2 * (step * BSTEP + t * 16));
                acc0[t] = __builtin_amdgcn_wmma_f32_16x16x4_f32(
                    false, a0, false, b, (short)0, acc0[t], false, false);
                acc1[t] = __builtin_amdgcn_wmma_f32_16x16x4_f32(
                    false, a1, false, b, (short)0, acc1[t], false, false);
            }
        }
        __syncthreads();
    }

    // ---- store: vgpr r of a tile holds (row_in_tile = hsel*8 + r, col = 16t+m)
    #pragma unroll
    for (int t = 0; t < 4; ++t)
        #pragma unroll
        for (int r = 0; r < 8; ++r) {
            const long g0 = base + wave * 32 + hsel * 8 + r;
            const long g1 = g0 + 16;
            if (g0 < (long)N) outLocal[g0 * CCH + t * 16 + m] = acc0[t][r];
            if (g1 < (long)N) outLocal[g1 * CCH + t * 16 + m] = acc1[t][r];
        }
}

// ---------------------------------------------------------------------------
// K1: F_input = LN(F @ W_pre); pw = (coords @ W_pos)*alpha; atomically
// accumulate Fcat = [F*cos, F*sin, F*pw] into aux_sum, +1 into counts.
// One wave per row; lane handles channels {l, l+32}.
// ---------------------------------------------------------------------------
__global__ __launch_bounds__(256) void elk_premix(
    const float* __restrict__ F, const int* __restrict__ coords,
    const int* __restrict__ aux_id,
    const float* __restrict__ Wpre, const float* __restrict__ gpre,
    const float* __restrict__ bpre, const float* __restrict__ Wpos,
    const float* __restrict__ alpha,
    float* __restrict__ Finput, float* __restrict__ auxSum,
    float* __restrict__ counts, int N)
{
    __shared__ float sW[CCH * CCH];
    __shared__ float sPos[3 * CCH];
    __shared__ float sAl[CCH], sG[CCH], sBi[CCH];
    __shared__ float sRow[8][CCH];

    const int tid = threadIdx.x;
    for (int i = tid; i < CCH * CCH; i += 256) sW[i] = Wpre[i];
    for (int i = tid; i < 3 * CCH; i += 256) sPos[i] = Wpos[i];
    if (tid < CCH) { sAl[tid] = alpha[tid]; sG[tid] = gpre[tid]; sBi[tid] = bpre[tid]; }
    __syncthreads();

    const int lane = tid & 31, wave = tid >> 5;
    const int c0 = lane, c1 = lane + 32;
    const long stride = (long)gridDim.x * 8;

    for (long row = (long)blockIdx.x * 8 + wave; row < (long)N; row += stride) {
        sRow[wave][c0] = F[row * CCH + c0];     // intra-wave LDS is in-order
        sRow[wave][c1] = F[row * CCH + c1];
        float a0 = 0.f, a1 = 0.f;
        #pragma unroll 8
        for (int j = 0; j < CCH; ++j) {
            const float f = sRow[wave][j];      // LDS broadcast read
            a0 = fmaf(f, sW[j * CCH + c0], a0);
            a1 = fmaf(f, sW[j * CCH + c1], a1);
        }
        float s = a0 + a1, q = a0 * a0 + a1 * a1;
        #pragma unroll
        for (int off = 16; off > 0; off >>= 1) {
            s += __shfl_xor(s, off, 32);
            q += __shfl_xor(q, off, 32);
        }
        const float mu  = s * (1.0f / CCH);
        const float var = q * (1.0f / CCH) - mu * mu;
        const float inv = rsqrtf(var + 1e-6f);
        const float f0 = (a0 - mu) * inv * sG[c0] + sBi[c0];
        const float f1 = (a1 - mu) * inv * sG[c1] + sBi[c1];
        Finput[row * CCH + c0] = f0;
        Finput[row * CCH + c1] = f1;

        const float x = (float)coords[row * 3 + 0];
        const float y = (float)coords[row * 3 + 1];
        const float z = (float)coords[row * 3 + 2];
        const float pw0 = (x * sPos[c0] + y * sPos[CCH + c0] + z * sPos[2 * CCH + c0]) * sAl[c0];
        const float pw1 = (x * sPos[c1] + y * sPos[CCH + c1] + z * sPos[2 * CCH + c1]) * sAl[c1];
        float sn0, cs0, sn1, cs1;
        sincosf(pw0, &sn0, &cs0);
        sincosf(pw1, &sn1, &cs1);

        float* dst = auxSum + (long)aux_id[row] * (3 * CCH);
        atomicAdd(dst + c0, f0 * cs0);
        atomicAdd(dst + c1, f1 * cs1);
        atomicAdd(dst + CCH + c0, f0 * sn0);
        atomicAdd(dst + CCH + c1, f1 * sn1);
        atomicAdd(dst + 2 * CCH + c0, f0 * pw0);
        atomicAdd(dst + 2 * CCH + c1, f1 * pw1);
        if (lane == 0) atomicAdd(counts + aux_id[row], 1.0f);
    }
}

// ---------------------------------------------------------------------------
// K3a: aux = aux_sum / max(counts, 1)
// ---------------------------------------------------------------------------
__global__ __launch_bounds__(256) void elk_aux_norm(
    float* __restrict__ auxSum, const float* __restrict__ counts, long total)
{
    const long i = (long)blockIdx.x * 256 + threadIdx.x;
    if (i < total) {
        const float c = counts[i / (3 * CCH)];
        auxSum[i] = auxSum[i] / fmaxf(c, 1.0f);
    }
}

// ---------------------------------------------------------------------------
// K3b: back[m] = sum_k aux[aux_nbr[m,k]]  (one block per aux row, 192 cols)
// ---------------------------------------------------------------------------
__global__ __launch_bounds__(192) void elk_aux_gather(
    const float* __restrict__ aux, const int* __restrict__ aux_nbr,
    float* __restrict__ back, int M)
{
    const int mrow = blockIdx.x;
    const int c = threadIdx.x;
    float s = 0.f;
    #pragma unroll 1
    for (int k = 0; k < KNB; ++k) {
        const int idx = aux_nbr[mrow * KNB + k];   // uniform -> scalar load
        s += aux[(long)idx * (3 * CCH) + c];
    }
    back[(long)mrow * (3 * CCH) + c] = s;
}

// ---------------------------------------------------------------------------
// K4: new_F = V1*cos + V2*sin + V3 - F_in*pw;  out = relu(LN(new_F)+LN(local))
// One wave per row; combined 4-value shuffle reduction for both LayerNorms.
// ---------------------------------------------------------------------------
__global__ __launch_bounds__(256) void elk_final(
    const float* __restrict__ Finput, const float* __restrict__ localF,
    const float* __restrict__ back, const int* __restrict__ aux_id,
    const int* __restrict__ coords, const float* __restrict__ Wpos,
    const float* __restrict__ alpha,
    const float* __restrict__ gnorm, const float* __restrict__ bnorm,
    const float* __restrict__ glocal, const float* __restrict__ blocal,
    float* __restrict__ out, int N)
{
    __shared__ float sPos[3 * CCH], sAl[CCH], sGn[CCH], sBn[CCH], sGl[CCH], sBl[CCH];
    const int tid = threadIdx.x;
    for (int i = tid; i < 3 * CCH; i += 256) sPos[i] = Wpos[i];
    if (tid < CCH) {
        sAl[tid] = alpha[tid];
        sGn[tid] = gnorm[tid];  sBn[tid] = bnorm[tid];
        sGl[tid] = glocal[tid]; sBl[tid] = blocal[tid];
    }
    __syncthreads();

    const int lane = tid & 31, wave = tid >> 5;
    const int c0 = lane, c1 = lane + 32;
    const long stride = (long)gridDim.x * 8;

    for (long row = (long)blockIdx.x * 8 + wave; row < (long)N; row += stride) {
        const float x = (float)coords[row * 3 + 0];
        const float y = (float)coords[row * 3 + 1];
        const float z = (float)coords[row * 3 + 2];
        const float pw0 = (x * sPos[c0] + y * sPos[CCH + c0] + z * sPos[2 * CCH + c0]) * sAl[c0];
        const float pw1 = (x * sPos[c1] + y * sPos[CCH + c1] + z * sPos[2 * CCH + c1]) * sAl[c1];
        float sn0, cs0, sn1, cs1;
        sincosf(pw0, &sn0, &cs0);
        sincosf(pw1, &sn1, &cs1);

        const float* V = back + (long)aux_id[row] * (3 * CCH);
        const float fi0 = Finput[row * CCH + c0];
        const float fi1 = Finput[row * CCH + c1];
        const float nf0 = V[c0] * cs0 + V[CCH + c0] * sn0 + V[2 * CCH + c0] - fi0 * pw0;
        const float nf1 = V[c1] * cs1 + V[CCH + c1] * sn1 + V[2 * CCH + c1] - fi1 * pw1;
        const float lo0 = localF[row * CCH + c0];
        const float lo1 = localF[row * CCH + c1];

        float s1 = nf0 + nf1, q1 = nf0 * nf0 + nf1 * nf1;
        float s2 = lo0 + lo1, q2 = lo0 * lo0 + lo1 * lo1;
        #pragma unroll
        for (int off = 16; off > 0; off >>= 1) {
            s1 += __shfl_xor(s1, off, 32); q1 += __shfl_xor(q1, off, 32);
            s2 += __shfl_xor(s2, off, 32); q2 += __shfl_xor(q2, off, 32);
        }
        const float mu1 = s1 * (1.f / CCH), var1 = q1 * (1.f / CCH) - mu1 * mu1;
        const float mu2 = s2 * (1.f / CCH), var2 = q2 * (1.f / CCH) - mu2 * mu2;
        const float i1 = rsqrtf(var1 + 1e-6f), i2 = rsqrtf(var2 + 1e-6f);
        const float o0 = fmaxf((nf0 - mu1) * i1 * sGn[c0] + sBn[c0] +
                               (lo0 - mu2) * i2 * sGl[c0] + sBl[c0], 0.f);
        const float o1 = fmaxf((nf1 - mu1) * i1 * sGn[c1] + sBn[c1] +
                               (lo1 - mu2) * i2 * sGl[c1] + sBl[c1], 0.f);
        out[row * CCH + c0] = o0;
        out[row * CCH + c1] = o1;
    }
}

// ---------------------------------------------------------------------------
extern "C" void kernel_launch(void* const* d_in, const int* in_sizes, int n_in,
                              void* d_out, int out_size, void* d_ws, size_t ws_size,
                              hipStream_t stream)
{
    const float* F       = (const float*)d_in[0];
    const int*   coords  = (const int*)d_in[1];
    const int*   nbr     = (const int*)d_in[2];
    const int*   aux_id  = (const int*)d_in[3];
    const int*   aux_nbr = (const int*)d_in[4];
    const float* Wpre    = (const float*)d_in[5];
    const float* gpre    = (const float*)d_in[6];
    const float* bpre    = (const float*)d_in[7];
    const float* Wconv   = (const float*)d_in[8];
    const float* Wpos    = (const float*)d_in[9];
    const float* alpha   = (const float*)d_in[10];
    const float* gnorm   = (const float*)d_in[11];
    const float* bnorm   = (const float*)d_in[12];
    const float* glocal  = (const float*)d_in[13];
    const float* blocal  = (const float*)d_in[14];
    (void)n_in; (void)out_size; (void)ws_size;

    const int N = in_sizes[0] / CCH;
    const int M = in_sizes[4] / KNB;

    // Workspace layout (~200 MB): F_input | local | aux_sum | counts | back
    char* ws = (char*)d_ws;
    float* Finput = (float*)ws;             size_t off  = (size_t)N * CCH * 4;
    float* localF = (float*)(ws + off);            off += (size_t)N * CCH * 4;
    float* auxSum = (float*)(ws + off);            off += (size_t)M * 3 * CCH * 4;
    float* counts = (float*)(ws + off);            off += (size_t)M * 4;
    float* back   = (float*)(ws + off);

    // zero only the atomic-accumulation region (aux_sum + counts, contiguous)
    hipMemsetAsync(auxSum, 0, (size_t)M * 3 * CCH * 4 + (size_t)M * 4, stream);

    elk_premix<<<1024, 256, 0, stream>>>(F, coords, aux_id, Wpre, gpre, bpre,
                                         Wpos, alpha, Finput, auxSum, counts, N);
    elk_conv_wmma<<<(N + 255) / 256, 256, 0, stream>>>(F, nbr, Wconv, localF, N);

    const long tot = (long)M * 3 * CCH;
    elk_aux_norm<<<(int)((tot + 255) / 256), 256, 0, stream>>>(auxSum, counts, tot);
    elk_aux_gather<<<M, 192, 0, stream>>>(auxSum, aux_nbr, back, M);

    elk_final<<<2048, 256, 0, stream>>>(Finput, localF, back, aux_id, coords,
                                        Wpos, alpha, gnorm, bnorm, glocal, blocal,
                                        (float*)d_out, N);
}